// ReceptiveHingeLoss_65395172049022
// MI455X (gfx1250) — compile-verified
//
#include <hip/hip_runtime.h>
#include <math.h>

// ---------------------------------------------------------------------------
// ReceptiveHingeLoss for MI455X (gfx1250, wave32, WMMA)
// Outputs: d_out[0] = loss, d_out[1 .. 1+1024*128) = kp1_desc (K,C) row-major
// ---------------------------------------------------------------------------

typedef float v2f __attribute__((ext_vector_type(2)));
typedef float v8f __attribute__((ext_vector_type(8)));

#define NN   8
#define CC   128
#define HR   32
#define WR   32
#define KP   1024
#define PP   (HR * WR)            // 1024 spatial positions per batch
#define POS_LAMBDA ((float)(374.359 * 0.3))
#define LOSS_DENOM ((float)(1024.0 * 374.359))

// --------------------------- 1) bilinear sampling ---------------------------
__global__ __launch_bounds__(128) void sample_desc_kernel(
    const float* __restrict__ kp1,    // [KP][4] : b, 0, y, x
    const float* __restrict__ desc1,  // [N][C][HR][WR]
    float* __restrict__ A,            // [KP][C]  (workspace, 16B aligned)
    float* __restrict__ out_desc)     // d_out + 1 (same layout)
{
  const int k = blockIdx.x;
  const int c = threadIdx.x;
  const float kb = kp1[k * 4 + 0];
  const float ky = kp1[k * 4 + 2];
  const float kx = kp1[k * 4 + 3];
  const int b = (int)kb;
  const float y = ky * 0.125f;       // / GRID_SIZE
  const float x = kx * 0.125f;
  float x0f = fminf(fmaxf(floorf(x), 0.0f), (float)(WR - 1));
  float y0f = fminf(fmaxf(floorf(y), 0.0f), (float)(HR - 1));
  float x1f = fminf(x0f + 1.0f, (float)(WR - 1));
  float y1f = fminf(y0f + 1.0f, (float)(HR - 1));
  const float wx = x - x0f;
  const float wy = y - y0f;
  const int x0 = (int)x0f, x1 = (int)x1f, y0 = (int)y0f, y1 = (int)y1f;
  const float* base = desc1 + ((size_t)b * CC + c) * HR * WR;
  const float d00 = base[y0 * WR + x0];
  const float d01 = base[y0 * WR + x1];
  const float d10 = base[y1 * WR + x0];
  const float d11 = base[y1 * WR + x1];
  const float r = d00 * (1.0f - wx) * (1.0f - wy)
                + d01 * wx * (1.0f - wy)
                + d10 * (1.0f - wx) * wy
                + d11 * wx * wy;
  A[k * CC + c] = r;
  out_desc[k * CC + c] = r;
}

// ------------------- 2) warped grid + visibility products -------------------
__global__ __launch_bounds__(256) void wgrid_vis_kernel(
    const float* __restrict__ homo21,   // [N][3][3]
    const float* __restrict__ vis_mask, // [N][1][256][256]
    float* __restrict__ wgrid,          // [N][PP][2]
    float* __restrict__ vis)            // [N][PP]
{
  const int idx = blockIdx.x * blockDim.x + threadIdx.x;
  if (idx >= NN * PP) return;
  const int n = idx >> 10;
  const int p = idx & (PP - 1);
  const int h = p >> 5;
  const int w = p & (WR - 1);
  const float px = (float)(w * 8 + 4);
  const float py = (float)(h * 8 + 4);
  const float* H = homo21 + n * 9;
  const float w0 = H[0] * px + H[1] * py + H[2];
  const float w1 = H[3] * px + H[4] * py + H[5];
  const float w2 = H[6] * px + H[7] * py + H[8];
  wgrid[idx * 2 + 0] = w0 / w2;
  wgrid[idx * 2 + 1] = w1 / w2;
  float prod = 1.0f;
  const float* vm = vis_mask + (size_t)n * 256 * 256;
  for (int i = 0; i < 8; ++i)
    for (int j = 0; j < 8; ++j)
      prod *= vm[(h * 8 + i) * 256 + (w * 8 + j)];
  vis[idx] = prod;
}

// -------- 3) dot_desc GEMM via V_WMMA_F32_16X16X4_F32 (exact f32 math) ------
// One wave computes a 16(kp) x 64(pos) tile for one n: 4 accumulators share
// the A fragment.  A: 16x4 f32 layout (lanes 0-15: K=0/1; 16-31: K=2/3);
// B mirrored; C/D: VGPR r -> row r + 8*hi, col = lane&15.
__global__ __launch_bounds__(32) void gemm_wmma_kernel(
    const float* __restrict__ A,      // [KP][CC]
    const float* __restrict__ desc2,  // [N][CC][PP]
    float* __restrict__ dot)          // [N][KP][PP]
{
  const int lane  = threadIdx.x;
  const int lh    = lane & 15;
  const int hi    = lane >> 4;        // 0 or 1
  const int ktile = blockIdx.x;       // 0..63
  const int n     = blockIdx.y;       // 0..7
  const int pgrp  = blockIdx.z;       // 0..15 -> 64 positions

  const float* B = desc2 + (size_t)n * CC * PP;
  const int k0 = ktile * 16 + lh;
  const int pbase = pgrp * 64;

  v8f acc0 = {}, acc1 = {}, acc2 = {}, acc3 = {};
  for (int cb = 0; cb < CC; cb += 4) {
    v2f a;
    a.x = A[k0 * CC + cb + 2 * hi + 0];
    a.y = A[k0 * CC + cb + 2 * hi + 1];
    const float* Brow0 = B + (size_t)(cb + 2 * hi + 0) * PP + pbase + lh;
    const float* Brow1 = B + (size_t)(cb + 2 * hi + 1) * PP + pbase + lh;
    v2f b0, b1, b2, b3;
    b0.x = Brow0[0];  b0.y = Brow1[0];
    b1.x = Brow0[16]; b1.y = Brow1[16];
    b2.x = Brow0[32]; b2.y = Brow1[32];
    b3.x = Brow0[48]; b3.y = Brow1[48];
    acc0 = __builtin_amdgcn_wmma_f32_16x16x4_f32(false, a, false, b0, (short)0, acc0, false, false);
    acc1 = __builtin_amdgcn_wmma_f32_16x16x4_f32(false, a, false, b1, (short)0, acc1, false, false);
    acc2 = __builtin_amdgcn_wmma_f32_16x16x4_f32(false, a, false, b2, (short)0, acc2, false, false);
    acc3 = __builtin_amdgcn_wmma_f32_16x16x4_f32(false, a, false, b3, (short)0, acc3, false, false);
  }
  float* Dn = dot + (size_t)n * KP * PP;
  for (int r = 0; r < 8; ++r) {
    const int kk = ktile * 16 + r + 8 * hi;
    float* row = Dn + (size_t)kk * PP + pbase + lh;
    row[0]  = acc0[r];
    row[16] = acc1[r];
    row[32] = acc2[r];
    row[48] = acc3[r];
  }
}

// ------------- 4) per-(n,k): s, separable Gaussian ns, hinge loss -----------
__global__ __launch_bounds__(256) void blur_loss_kernel(
    const float* __restrict__ kp1,     // [KP][4]
    const float* __restrict__ dot,     // [N][KP][PP]
    const float* __restrict__ wgrid,   // [N][PP][2]
    const float* __restrict__ vis,     // [N][PP]
    float* __restrict__ partial)       // [N*KP]
{
  __shared__ float s_img[PP];
  __shared__ float t0[PP];
  __shared__ float t1[PP];
  __shared__ float t2[PP];
  __shared__ float wk[48];             // 0..30: ks31 s7 | 31..41: ks11 s4 | 42..46: ks5 s2
  __shared__ float red[256];

  const int k = blockIdx.x;            // keypoint
  const int n = blockIdx.y;            // batch
  const int tid = threadIdx.x;

  if (tid == 0) {
    float sum = 0.0f;
    for (int i = 0; i < 31; ++i) { float x = (float)i - 15.0f; float v = expf(-(x * x) / (2.0f * 7.0f * 7.0f)); wk[i] = v; sum += v; }
    for (int i = 0; i < 31; ++i) wk[i] /= sum;
    sum = 0.0f;
    for (int i = 0; i < 11; ++i) { float x = (float)i - 5.0f; float v = expf(-(x * x) / (2.0f * 4.0f * 4.0f)); wk[31 + i] = v; sum += v; }
    for (int i = 0; i < 11; ++i) wk[31 + i] /= sum;
    sum = 0.0f;
    for (int i = 0; i < 5; ++i) { float x = (float)i - 2.0f; float v = expf(-(x * x) / (2.0f * 2.0f * 2.0f)); wk[42 + i] = v; sum += v; }
    for (int i = 0; i < 5; ++i) wk[42 + i] /= sum;
  }

  const float kpy = kp1[k * 4 + 2];
  const float kpx = kp1[k * 4 + 3];
  const float* wg = wgrid + (size_t)n * PP * 2;

  for (int q = 0; q < 4; ++q) {
    const int cid = tid + q * 256;
    const float gx = wg[cid * 2 + 0];
    const float gy = wg[cid * 2 + 1];
    const float dx = kpx - gx;
    const float dy = kpy - gy;
    s_img[cid] = (dx * dx + dy * dy <= 56.25f) ? 1.0f : 0.0f;   // (8-0.5)^2
  }
  __syncthreads();

  // pass 1: 1-D conv along h (zero padded)
  for (int q = 0; q < 4; ++q) {
    const int cid = tid + q * 256;
    const int h = cid >> 5, w = cid & 31;
    float a0 = 0.0f, a1 = 0.0f, a2 = 0.0f;
    for (int j = 0; j < 31; ++j) { int hh = h + j - 15; if (hh >= 0 && hh < HR) a0 += wk[j]      * s_img[hh * WR + w]; }
    for (int j = 0; j < 11; ++j) { int hh = h + j - 5;  if (hh >= 0 && hh < HR) a1 += wk[31 + j] * s_img[hh * WR + w]; }
    for (int j = 0; j < 5;  ++j) { int hh = h + j - 2;  if (hh >= 0 && hh < HR) a2 += wk[42 + j] * s_img[hh * WR + w]; }
    t0[cid] = a0; t1[cid] = a1; t2[cid] = a2;
  }
  __syncthreads();

  // pass 2: 1-D conv along w, combine, fuse hinge loss
  const float* Drow = dot + ((size_t)n * KP + k) * PP;
  const float* vn = vis + (size_t)n * PP;
  float lsum = 0.0f;
  for (int q = 0; q < 4; ++q) {
    const int cid = tid + q * 256;
    const int h = cid >> 5, w = cid & 31;
    float a0 = 0.0f, a1 = 0.0f, a2 = 0.0f;
    for (int j = 0; j < 31; ++j) { int ww = w + j - 15; if (ww >= 0 && ww < WR) a0 += wk[j]      * t0[h * WR + ww]; }
    for (int j = 0; j < 11; ++j) { int ww = w + j - 5;  if (ww >= 0 && ww < WR) a1 += wk[31 + j] * t1[h * WR + ww]; }
    for (int j = 0; j < 5;  ++j) { int ww = w + j - 2;  if (ww >= 0 && ww < WR) a2 += wk[42 + j] * t2[h * WR + ww]; }
    const float sv = s_img[cid];
    const float ns = (a0 + a1 + a2) - 3.0f * sv;
    const float v  = vn[cid];
    const float s_m  = sv * v;
    const float ns_m = ns * v;
    const float D = Drow[cid];
    const float pos = fmaxf(1.0f - D, 0.0f);
    const float neg = fmaxf(D - 0.2f, 0.0f);
    lsum += POS_LAMBDA * s_m * pos + ns_m * neg;
  }
  red[tid] = lsum;
  __syncthreads();
  for (int s = 128; s > 0; s >>= 1) {
    if (tid < s) red[tid] += red[tid + s];
    __syncthreads();
  }
  if (tid == 0) partial[n * KP + k] = red[0];
}

// --------------------- 5) deterministic final reduction ---------------------
__global__ __launch_bounds__(256) void reduce_kernel(
    const float* __restrict__ partial,  // [8192]
    float* __restrict__ out)            // d_out[0]
{
  __shared__ float red[256];
  const int tid = threadIdx.x;
  float s = 0.0f;
  for (int i = 0; i < 32; ++i) s += partial[tid + 256 * i];  // fixed order
  red[tid] = s;
  __syncthreads();
  for (int st = 128; st > 0; st >>= 1) {
    if (tid < st) red[tid] += red[tid + st];
    __syncthreads();
  }
  if (tid == 0) out[0] = red[0] / LOSS_DENOM;
}

// ---------------------------------------------------------------------------
extern "C" void kernel_launch(void* const* d_in, const int* in_sizes, int n_in,
                              void* d_out, int out_size, void* d_ws, size_t ws_size,
                              hipStream_t stream) {
  const float* kp1   = (const float*)d_in[0];  // [1024][4]
  const float* desc1 = (const float*)d_in[1];  // [8][128][32][32]
  const float* desc2 = (const float*)d_in[2];  // [8][128][32][32]
  const float* homo  = (const float*)d_in[3];  // [8][3][3]
  const float* vism  = (const float*)d_in[4];  // [8][1][256][256]
  float* out = (float*)d_out;                  // [1 + 1024*128]

  float* ws    = (float*)d_ws;
  float* dot   = ws;                                  // 8*1024*1024
  float* A     = dot + (size_t)NN * KP * PP;          // 1024*128
  float* wgrid = A + KP * CC;                         // 8*1024*2
  float* vis   = wgrid + NN * PP * 2;                 // 8*1024
  float* part  = vis + NN * PP;                       // 8192

  sample_desc_kernel<<<KP, CC, 0, stream>>>(kp1, desc1, A, out + 1);
  wgrid_vis_kernel<<<(NN * PP + 255) / 256, 256, 0, stream>>>(homo, vism, wgrid, vis);
  gemm_wmma_kernel<<<dim3(KP / 16, NN, 16), 32, 0, stream>>>(A, desc2, dot);
  blur_loss_kernel<<<dim3(KP, NN), 256, 0, stream>>>(kp1, dot, wgrid, vis, part);
  reduce_kernel<<<1, 256, 0, stream>>>(part, out);
}